// RecurrentRGCN_19696720020162
// MI455X (gfx1250) — compile-verified
//
#include <hip/hip_runtime.h>
#include <math.h>

// Problem constants
#define HDIM 128
#define KTOT 384          // concatenated K: [norm*agg | h | has_in?h:0]
#define ROWPAD 388        // LDS row stride in floats: 388%64=4 -> conflict-free frag reads
#define MBLK 32           // node rows per workgroup (2 M-tiles per wave)
#define RRELU_SLOPE 0.2291666666666667f   // (1/8 + 1/3)/2 = 11/48

typedef float v2f __attribute__((ext_vector_type(2)));
typedef float v8f __attribute__((ext_vector_type(8)));

__device__ __forceinline__ void atomAddF(float* p, float v) {
  unsafeAtomicAdd(p, v);   // native global_atomic_add_f32 on gfx1250
}

// ---------------- in-degree ----------------
__global__ __launch_bounds__(256)
void k_deg(const int* __restrict__ dst, float* __restrict__ deg, int nedges) {
  int e = blockIdx.x * 256 + threadIdx.x;
  if (e < nedges) atomAddF(&deg[dst[e]], 1.0f);
}

// ---------------- build concatenated weights: [Wn ; We ; Wl - We] per layer ----------------
__global__ __launch_bounds__(256)
void k_wcat(const float* __restrict__ wn, const float* __restrict__ wl,
            const float* __restrict__ we, float* __restrict__ wcat) {
  int l = blockIdx.y;
  int idx = blockIdx.x * 256 + threadIdx.x;          // 0 .. 384*128-1
  if (idx >= KTOT * HDIM) return;
  int k = idx >> 7;                                  // row (input dim)
  int base = l * HDIM * HDIM;
  float v;
  if (k < HDIM)            v = wn[base + idx];
  else if (k < 2 * HDIM)   v = we[base + idx - HDIM * HDIM];
  else                     v = wl[base + idx - 2 * HDIM * HDIM]
                             - we[base + idx - 2 * HDIM * HDIM];
  wcat[l * KTOT * HDIM + idx] = v;
}

// ---------------- edge aggregation: agg[dst] += h[src] + rel[etype] ----------------
// one wave (32 lanes) per edge, 4 floats/lane, fp32 atomics land in L2 (agg fits in 192MB L2)
__global__ __launch_bounds__(256)
void k_edge(const float* __restrict__ h, const float* __restrict__ rel,
            const int* __restrict__ src, const int* __restrict__ dst,
            const int* __restrict__ ety, float* __restrict__ agg, int nedges) {
  int e = blockIdx.x * 8 + (threadIdx.x >> 5);
  if (e >= nedges) return;
  int lane = threadIdx.x & 31;
  int s = src[e], d = dst[e], t = ety[e];
  float4 a = ((const float4*)(h   + (size_t)s * HDIM))[lane];
  float4 b = ((const float4*)(rel + (size_t)t * HDIM))[lane];
  float* out = agg + (size_t)d * HDIM + lane * 4;
  atomAddF(out + 0, a.x + b.x);
  atomAddF(out + 1, a.y + b.y);
  atomAddF(out + 2, a.z + b.z);
  atomAddF(out + 3, a.w + b.w);
}

// ---------------- per-node layer GEMM via V_WMMA_F32_16X16X4_F32 ----------------
// Workgroup = 256 threads = 8 waves, owns MBLK=32 node rows. Each wave: one 16-col
// tile x two 16-row tiles -> two independent accumulators sharing every B fragment
// (halves B traffic per WMMA, and the 2 WMMAs per k-step are independent -> deeper
// pipelining than a pure C-chain). A block (32x384) staged in LDS via float4.
// Output overwrites `agg` rows in place (exclusive ownership per block).
__global__ __launch_bounds__(256)
void k_gemm(const float* __restrict__ hprev,
            float* __restrict__ agg,                  // in: edge sums; out: h_next
            const float* __restrict__ deg,
            const float* __restrict__ wcat,           // 384x128 row-major
            int nents) {
  __shared__ float lds_a[MBLK * ROWPAD];              // 32*388*4 = 48.5 KB
  const int base = blockIdx.x * MBLK;
  const int tid = threadIdx.x;

  // ---- stage A (float4 granularity): row r, segment s: 0 = norm*agg, 1 = h, 2 = has_in?h:0
  // 32 rows x 96 float4s = 3072 float4s, 12 per thread
  for (int idx = tid; idx < MBLK * (KTOT / 4); idx += 256) {
    int r = idx / (KTOT / 4);
    int c = idx - r * (KTOT / 4);                     // 0..95
    int s = c >> 5;                                   // segment 0..2
    int k4 = c & 31;                                  // float4 index within 128-wide segment
    int node = base + r;
    float4 v = make_float4(0.f, 0.f, 0.f, 0.f);
    if (node < nents) {
      float dg = deg[node];
      if (s == 0) {
        float nrm = dg > 0.0f ? 1.0f / dg : 0.0f;
        float4 a = ((const float4*)(agg + (size_t)node * HDIM))[k4];
        v = make_float4(a.x * nrm, a.y * nrm, a.z * nrm, a.w * nrm);
      } else if (s == 1) {
        v = ((const float4*)(hprev + (size_t)node * HDIM))[k4];
      } else if (dg > 0.0f) {
        v = ((const float4*)(hprev + (size_t)node * HDIM))[k4];
      }
    }
    // (r*388 + s*128 + 4*k4)*4 bytes is 16B aligned -> ds_store_b128
    ((float4*)&lds_a[r * ROWPAD + s * HDIM + k4 * 4])[0] = v;
  }
  __syncthreads();

  const int wave = tid >> 5;
  const int lane = tid & 31;
  const int half = lane >> 4;        // ISA: lanes 16-31 carry K+2/K+3 (A and B)
  const int lmod = lane & 15;
  const int col  = wave * 16 + lmod;

  v8f acc0 = {0.f, 0.f, 0.f, 0.f, 0.f, 0.f, 0.f, 0.f};
  v8f acc1 = {0.f, 0.f, 0.f, 0.f, 0.f, 0.f, 0.f, 0.f};
#if __has_builtin(__builtin_amdgcn_wmma_f32_16x16x4_f32)
  const float* arow0 = &lds_a[lmod * ROWPAD + 2 * half];            // rows 0-15
  const float* arow1 = &lds_a[(16 + lmod) * ROWPAD + 2 * half];     // rows 16-31
  const float* bcol  = wcat + (2 * half) * HDIM + col;
  #pragma unroll 8
  for (int kk = 0; kk < KTOT; kk += 4) {
    v2f bf;
    bf.x = bcol[kk * HDIM];
    bf.y = bcol[kk * HDIM + HDIM];
    v2f a0, a1;
    a0.x = arow0[kk]; a0.y = arow0[kk + 1];
    a1.x = arow1[kk]; a1.y = arow1[kk + 1];
    acc0 = __builtin_amdgcn_wmma_f32_16x16x4_f32(false, a0, false, bf,
                                                 (short)0, acc0, false, false);
    acc1 = __builtin_amdgcn_wmma_f32_16x16x4_f32(false, a1, false, bf,
                                                 (short)0, acc1, false, false);
  }
#else
  for (int r = 0; r < 8; ++r) {
    int row = r + 8 * half;
    float s0 = 0.0f, s1 = 0.0f;
    for (int k = 0; k < KTOT; ++k) {
      float w = wcat[k * HDIM + col];
      s0 += lds_a[row * ROWPAD + k] * w;
      s1 += lds_a[(row + 16) * ROWPAD + k] * w;
    }
    acc0[r] = s0; acc1[r] = s1;
  }
#endif

  // D layout: VGPR r -> row (8*half + r) within tile; apply rrelu, store
  if (base + MBLK <= nents) {                         // fast path: no guards
    #pragma unroll
    for (int r = 0; r < 8; ++r) {
      int row = base + 8 * half + r;
      float v0 = acc0[r];
      v0 = v0 >= 0.0f ? v0 : v0 * RRELU_SLOPE;
      agg[(size_t)row * HDIM + col] = v0;
      float v1 = acc1[r];
      v1 = v1 >= 0.0f ? v1 : v1 * RRELU_SLOPE;
      agg[(size_t)(row + 16) * HDIM + col] = v1;
    }
  } else {
    #pragma unroll
    for (int r = 0; r < 8; ++r) {
      int row = base + 8 * half + r;
      if (row < nents) {
        float v0 = acc0[r];
        v0 = v0 >= 0.0f ? v0 : v0 * RRELU_SLOPE;
        agg[(size_t)row * HDIM + col] = v0;
      }
      if (row + 16 < nents) {
        float v1 = acc1[r];
        v1 = v1 >= 0.0f ? v1 : v1 * RRELU_SLOPE;
        agg[(size_t)(row + 16) * HDIM + col] = v1;
      }
    }
  }
}

// ---------------- scalar gate: sum_n (h[n].rg_w + rg_b) * rg1_w[n] ----------------
__global__ __launch_bounds__(256)
void k_red(const float* __restrict__ h, const float* __restrict__ rg_w,
           const float* __restrict__ rg_b, const float* __restrict__ rg1_w,
           float* __restrict__ acc, int nents) {
  int n = blockIdx.x * 8 + (threadIdx.x >> 5);
  int lane = threadIdx.x & 31;
  float part = 0.0f;
  if (n < nents) {
    float4 a = ((const float4*)(h + (size_t)n * HDIM))[lane];
    float4 b = ((const float4*)rg_w)[lane];
    part = a.x * b.x + a.y * b.y + a.z * b.z + a.w * b.w;
  }
  for (int off = 16; off > 0; off >>= 1) part += __shfl_down(part, off, 32);
  if (lane == 0 && n < nents) atomAddF(acc, (part + rg_b[0]) * rg1_w[n]);
}

// ---------------- final blend: out = h*sigmoid(s) + emb*(1-sigmoid(s)) ----------------
__global__ __launch_bounds__(256)
void k_fin(const float* __restrict__ h, const float* __restrict__ emb,
           const float* __restrict__ acc, const float* __restrict__ rg1_b,
           float* __restrict__ out, int total) {
  int i = blockIdx.x * 256 + threadIdx.x;
  if (i >= total) return;
  float s = acc[0] + rg1_b[0];
  float w = 1.0f / (1.0f + __expf(-s));
  out[i] = h[i] * w + emb[i] * (1.0f - w);
}

extern "C" void kernel_launch(void* const* d_in, const int* in_sizes, int n_in,
                              void* d_out, int out_size, void* d_ws, size_t ws_size,
                              hipStream_t stream) {
  const float* emb_ent = (const float*)d_in[0];
  const float* emb_rel = (const float*)d_in[1];
  const float* w_nb    = (const float*)d_in[2];
  const float* w_lp    = (const float*)d_in[3];
  const float* w_ev    = (const float*)d_in[4];
  const float* rg_w    = (const float*)d_in[5];
  const float* rg_b    = (const float*)d_in[6];
  const float* rg1_w   = (const float*)d_in[7];
  const float* rg1_b   = (const float*)d_in[8];
  const int*   src     = (const int*)d_in[9];
  const int*   dst     = (const int*)d_in[10];
  const int*   ety     = (const int*)d_in[11];
  float* out = (float*)d_out;

  const int nents  = in_sizes[0] / HDIM;   // 100000
  const int nedges = in_sizes[9];          // 1000000

  // workspace layout (floats): bufA | bufB | deg | wcat(2x384x128) | acc
  float* bufA = (float*)d_ws;
  float* bufB = bufA + (size_t)nents * HDIM;
  float* deg  = bufB + (size_t)nents * HDIM;
  float* wcat = deg + nents;
  float* acc  = wcat + 2 * KTOT * HDIM;

  hipMemsetAsync(deg, 0, sizeof(float) * (size_t)nents, stream);
  hipMemsetAsync(acc, 0, sizeof(float), stream);
  hipMemsetAsync(bufA, 0, sizeof(float) * (size_t)nents * HDIM, stream);
  hipMemsetAsync(bufB, 0, sizeof(float) * (size_t)nents * HDIM, stream);

  k_deg<<<(nedges + 255) / 256, 256, 0, stream>>>(dst, deg, nedges);
  dim3 wg((KTOT * HDIM + 255) / 256, 2);
  k_wcat<<<wg, 256, 0, stream>>>(w_nb, w_lp, w_ev, wcat);

  // layer 0: h = emb_ent -> bufA
  k_edge<<<(nedges + 7) / 8, 256, 0, stream>>>(emb_ent, emb_rel, src, dst, ety, bufA, nedges);
  k_gemm<<<(nents + MBLK - 1) / MBLK, 256, 0, stream>>>(emb_ent, bufA, deg, wcat, nents);

  // layer 1: h = bufA -> bufB
  k_edge<<<(nedges + 7) / 8, 256, 0, stream>>>(bufA, emb_rel, src, dst, ety, bufB, nedges);
  k_gemm<<<(nents + MBLK - 1) / MBLK, 256, 0, stream>>>(bufA, bufB, deg, wcat + KTOT * HDIM, nents);

  // reset-gate scalar + final blend
  k_red<<<(nents + 7) / 8, 256, 0, stream>>>(bufB, rg_w, rg_b, rg1_w, acc, nents);
  int total = nents * HDIM;
  k_fin<<<(total + 255) / 256, 256, 0, stream>>>(bufB, emb_ent, acc, rg1_b, out, total);
}